// GraphTransformerBlock_5677946765952
// MI455X (gfx1250) — compile-verified
//
#include <hip/hip_runtime.h>
#include <hip/hip_bf16.h>
#include <math.h>

typedef __attribute__((ext_vector_type(2))) float v2f;
typedef __attribute__((ext_vector_type(8))) float v8f;
typedef int b128_t __attribute__((vector_size(16)));   // matches builtin param type

#define HD 8    // heads
#define OD 16   // per-head out channels
#define DIM 128

#if __has_builtin(__builtin_amdgcn_global_load_async_to_lds_b128) && \
    __has_builtin(__builtin_amdgcn_s_wait_asynccnt)
#define HAS_ASYNC_LDS 1
#else
#define HAS_ASYNC_LDS 0
#endif

// ---------------- float atomic max via int trick ----------------
__device__ inline void atomicMaxFloatBits(int* addr, float val) {
  if (val >= 0.0f) {
    atomicMax(addr, __float_as_int(val));
  } else {
    atomicMin((unsigned int*)addr, __float_as_uint(val));
  }
}

// ---------------- cooperative 16 x K tile load into LDS -----------------------
// Uses CDNA5 GLOBAL_LOAD_ASYNC_TO_LDS_B128 (ASYNCcnt) when available; falls back
// to VGPR-bounce float4 copies otherwise. LDS rows padded to LDSK = K+4 floats
// (528B/2064B stride: 16B-aligned, conflict-free b64 fragment reads).
template <int K, int LDSK>
__device__ inline void load_tile_lds(const float* __restrict__ A, float* sA,
                                     int r0, int M, int tid) {
  for (int i = tid; i < 16 * (K / 4); i += 256) {
    int m = i / (K / 4);
    int kq = i - m * (K / 4);
    int row = r0 + m; if (row >= M) row = M - 1;
    const float* gp = A + (size_t)row * K + kq * 4;
    float* lp = sA + m * LDSK + kq * 4;
#if HAS_ASYNC_LDS
    __builtin_amdgcn_global_load_async_to_lds_b128(
        (__attribute__((address_space(1))) b128_t*)gp,
        (__attribute__((address_space(3))) b128_t*)lp, 0, 0);
#else
    *(float4*)lp = *(const float4*)gp;
#endif
  }
#if HAS_ASYNC_LDS
  __builtin_amdgcn_s_wait_asynccnt(0);
#endif
  __syncthreads();
}

// ---------------- init: zero accum/denom, -inf lmax ----------------
__global__ void init_kernel(float* __restrict__ accum, float* __restrict__ denom,
                            int* __restrict__ lmax_bits, int n_nodes) {
  int i = blockIdx.x * blockDim.x + threadIdx.x;
  int total = n_nodes * DIM;
  if (i < total) accum[i] = 0.0f;
  if (i < n_nodes * HD) {
    denom[i] = 0.0f;
    lmax_bits[i] = 0xFF800000; // -inf
  }
}

// ---------------- generic WMMA f32 GEMM:  C = act(A[MxK] @ W[KxNc] + bias) --------
// block: 256 threads (8 waves); 16 rows x 128 cols per block; one 16x16 tile per
// wave via V_WMMA_F32_16X16X4_F32 over K/4 steps.
template <int K, int ACT>
__global__ __launch_bounds__(256)
void gemm_wmma(const float* __restrict__ A, const float* __restrict__ W,
               const float* __restrict__ bias, float* __restrict__ C,
               int M, int Nc) {
  constexpr int LDSK = K + 4;
  __shared__ float sA[16 * LDSK];
  const int tid  = threadIdx.x;
  const int wave = tid >> 5;
  const int lane = tid & 31;
  const int r0   = blockIdx.x * 16;
  const int nb   = blockIdx.y * 128 + wave * 16;

  load_tile_lds<K, LDSK>(A, sA, r0, M, tid);

  const int n = lane & 15;   // column within tile / B lane
  const int t = lane >> 4;   // half-wave selector (K split / M split for C)
  v8f c = {};
#pragma unroll
  for (int kb = 0; kb < K; kb += 4) {
    // A fragment: lane (m = lane&15) holds K = kb+2t, kb+2t+1
    float2 av = *(const float2*)(sA + n * LDSK + kb + 2 * t);
    v2f a; a[0] = av.x; a[1] = av.y;
    // B fragment: lane n holds W[kb+2t][col], W[kb+2t+1][col]
    const float* wp = W + (size_t)(kb + 2 * t) * Nc + nb + n;
    v2f b; b[0] = wp[0]; b[1] = wp[Nc];
    c = __builtin_amdgcn_wmma_f32_16x16x4_f32(false, a, false, b, (short)0, c,
                                              false, false);
  }
  // epilogue: VGPR r holds row (t*8 + r), col = nb + n
#pragma unroll
  for (int r = 0; r < 8; ++r) {
    int row = r0 + t * 8 + r;
    if (row < M) {
      int col = nb + n;
      float v = c[r] + (bias ? bias[col] : 0.0f);
      if (ACT == 1) v = 0.5f * v * (1.0f + erff(v * 0.70710678118f)); // exact gelu
      C[(size_t)row * Nc + col] = v;
    }
  }
}

// ---------------- fused edge GEMM + logits + segment max -----------------------
// ee_tile = edge_attr[16 edges] @ W_e  (WMMA), then per wave (= head h):
// m = leaky_relu(ee + xl[src] + xr[dst]); logit = sum_o m*att[h,o]; atomicMax lmax.
__global__ __launch_bounds__(256)
void edge_logits_kernel(const float* __restrict__ edge_attr,
                        const float* __restrict__ We,
                        const float* __restrict__ xl, const float* __restrict__ xr,
                        const int* __restrict__ src, const int* __restrict__ dst,
                        const float* __restrict__ att,
                        float* __restrict__ logits, int* __restrict__ lmax_bits,
                        int E_) {
  constexpr int K = DIM, LDSK = K + 4;
  __shared__ float sA[16 * LDSK];
  const int tid  = threadIdx.x;
  const int wave = tid >> 5;       // == head h
  const int lane = tid & 31;
  const int e0   = blockIdx.x * 16;

  load_tile_lds<K, LDSK>(edge_attr, sA, e0, E_, tid);

  const int n = lane & 15;         // o index within head
  const int t = lane >> 4;
  const int nb = wave * 16;        // column base == head * O
  v8f c = {};
#pragma unroll
  for (int kb = 0; kb < K; kb += 4) {
    float2 av = *(const float2*)(sA + n * LDSK + kb + 2 * t);
    v2f a; a[0] = av.x; a[1] = av.y;
    const float* wp = We + (size_t)(kb + 2 * t) * DIM + nb + n;
    v2f b; b[0] = wp[0]; b[1] = wp[DIM];
    c = __builtin_amdgcn_wmma_f32_16x16x4_f32(false, a, false, b, (short)0, c,
                                              false, false);
  }
  const float aw = att[wave * OD + n];
#pragma unroll
  for (int r = 0; r < 8; ++r) {
    int e = e0 + t * 8 + r;
    if (e < E_) {
      int s_ = src[e], d_ = dst[e];
      float v = c[r] + xl[(size_t)s_ * DIM + nb + n] + xr[(size_t)d_ * DIM + nb + n];
      v = v > 0.0f ? v : 0.2f * v;                 // leaky_relu
      float p = v * aw;
      p += __shfl_xor(p, 1); p += __shfl_xor(p, 2); // reduce over 16 lanes (o)
      p += __shfl_xor(p, 4); p += __shfl_xor(p, 8);
      if (n == 0) {
        logits[(size_t)e * HD + wave] = p;
        atomicMaxFloatBits(&lmax_bits[d_ * HD + wave], p);
      }
    }
  }
}

// ---------------- softmax pass 2: exp + segment sum ----------------
__global__ void softmax_norm_kernel(const float* __restrict__ logits,
                                    const float* __restrict__ lmax,
                                    const int* __restrict__ dst,
                                    float* __restrict__ exl, float* __restrict__ denom,
                                    int E_) {
  int i = blockIdx.x * blockDim.x + threadIdx.x;
  if (i >= E_ * HD) return;
  int e = i >> 3, h = i & 7;
  int d_ = dst[e];
  float ex = __expf(logits[i] - lmax[d_ * HD + h]);
  exl[i] = ex;
  atomicAdd(&denom[d_ * HD + h], ex);
}

// ---------------- softmax pass 3: alpha-weighted scatter ----------------
// one wave32 per edge; lane owns 4 contiguous channels (coalesced float4 gather
// of xl[src], indices/alpha loaded once instead of per-channel).
__global__ __launch_bounds__(256)
void scatter_kernel(const float* __restrict__ xl,
                    const float* __restrict__ exl,
                    const float* __restrict__ denom,
                    const int* __restrict__ src, const int* __restrict__ dst,
                    float* __restrict__ accum, int E_) {
  int e = blockIdx.x * 8 + (threadIdx.x >> 5);
  if (e >= E_) return;
  int lane = threadIdx.x & 31;
  int h = lane >> 2;                       // 4 lanes per head (O=16, 4 ch/lane)
  int s_ = src[e], d_ = dst[e];
  float alpha = exl[(size_t)e * HD + h] / (denom[d_ * HD + h] + 1e-16f);
  float4 v = *(const float4*)(xl + (size_t)s_ * DIM + lane * 4);
  float* ap = accum + (size_t)d_ * DIM + lane * 4;
  atomicAdd(ap + 0, v.x * alpha);
  atomicAdd(ap + 1, v.y * alpha);
  atomicAdd(ap + 2, v.z * alpha);
  atomicAdd(ap + 3, v.w * alpha);
}

// ---------------- LayerNorm: out = LN(a + b (+ addv)) * g + be -----------------
// one wave32 per node, 4 floats per lane
__global__ __launch_bounds__(256)
void ln_kernel(const float* __restrict__ a, const float* __restrict__ b,
               const float* __restrict__ addv, const float* __restrict__ g,
               const float* __restrict__ be, float* __restrict__ out, int n_nodes) {
  int node = blockIdx.x * 8 + (threadIdx.x >> 5);
  if (node >= n_nodes) return;
  int lane = threadIdx.x & 31;
  size_t base = (size_t)node * DIM + lane * 4;
  float4 va = *(const float4*)(a + base);
  float4 vb = *(const float4*)(b + base);
  float v[4] = {va.x + vb.x, va.y + vb.y, va.z + vb.z, va.w + vb.w};
  if (addv) {
#pragma unroll
    for (int j = 0; j < 4; ++j) v[j] += addv[lane * 4 + j];
  }
  float s = v[0] + v[1] + v[2] + v[3];
  float sq = v[0] * v[0] + v[1] * v[1] + v[2] * v[2] + v[3] * v[3];
#pragma unroll
  for (int off = 16; off > 0; off >>= 1) {
    s  += __shfl_xor(s, off);
    sq += __shfl_xor(sq, off);
  }
  float mu = s * (1.0f / DIM);
  float var = sq * (1.0f / DIM) - mu * mu;
  float rstd = rsqrtf(var + 1e-5f);
  float4 vo;
  float* vp = &vo.x;
#pragma unroll
  for (int j = 0; j < 4; ++j)
    vp[j] = (v[j] - mu) * rstd * g[lane * 4 + j] + be[lane * 4 + j];
  *(float4*)(out + base) = vo;
}

// ---------------- launch ----------------
extern "C" void kernel_launch(void* const* d_in, const int* in_sizes, int n_in,
                              void* d_out, int out_size, void* d_ws, size_t ws_size,
                              hipStream_t stream) {
  const float* x         = (const float*)d_in[0];
  const int*   eidx      = (const int*)d_in[1];
  const float* edge_attr = (const float*)d_in[2];
  const float* W_l  = (const float*)d_in[4];
  const float* b_l  = (const float*)d_in[5];
  const float* W_r  = (const float*)d_in[6];
  const float* b_r  = (const float*)d_in[7];
  const float* W_e  = (const float*)d_in[8];
  const float* att  = (const float*)d_in[9];
  const float* b_att= (const float*)d_in[10];
  const float* W1   = (const float*)d_in[11];
  const float* b1   = (const float*)d_in[12];
  const float* W2   = (const float*)d_in[13];
  const float* b2   = (const float*)d_in[14];
  const float* g1   = (const float*)d_in[15];
  const float* be1  = (const float*)d_in[16];
  const float* g2   = (const float*)d_in[17];
  const float* be2  = (const float*)d_in[18];

  const int N = in_sizes[0] / DIM;
  const int E = in_sizes[2] / DIM;
  const int* src = eidx;
  const int* dst = eidx + E;

  float* ws     = (float*)d_ws;
  float* xl     = ws;
  float* xr     = xl    + (size_t)N * DIM;
  float* accum  = xr    + (size_t)N * DIM;
  float* hbuf   = accum + (size_t)N * DIM;
  float* t1     = hbuf  + (size_t)N * DIM;
  float* t2     = t1    + (size_t)N * 4 * DIM;
  float* logits = t2    + (size_t)N * DIM;
  float* exl    = logits + (size_t)E * HD;
  float* lmax   = exl    + (size_t)E * HD;   // float-bits, atomic max via int view
  float* denom  = lmax   + (size_t)N * HD;
  float* outp   = (float*)d_out;

  // 1) init accumulators
  init_kernel<<<(N * DIM + 255) / 256, 256, 0, stream>>>(accum, denom, (int*)lmax, N);
  // 2) xl = x@W_l + b_l ; xr = x@W_r + b_r
  dim3 gN((N + 15) / 16, 1);
  gemm_wmma<DIM, 0><<<gN, 256, 0, stream>>>(x, W_l, b_l, xl, N, DIM);
  gemm_wmma<DIM, 0><<<gN, 256, 0, stream>>>(x, W_r, b_r, xr, N, DIM);
  // 3) fused edge GEMM + logits + segment max
  edge_logits_kernel<<<(E + 15) / 16, 256, 0, stream>>>(
      edge_attr, W_e, xl, xr, src, dst, att, logits, (int*)lmax, E);
  // 4) exp + segment sum
  softmax_norm_kernel<<<(E * HD + 255) / 256, 256, 0, stream>>>(
      logits, lmax, dst, exl, denom, E);
  // 5) alpha-weighted scatter-add (wave per edge)
  scatter_kernel<<<(E + 7) / 8, 256, 0, stream>>>(
      xl, exl, denom, src, dst, accum, E);
  // 6) h = LN(x + accum + b_att)
  ln_kernel<<<(N + 7) / 8, 256, 0, stream>>>(x, accum, b_att, g1, be1, hbuf, N);
  // 7) t1 = gelu(h@W1 + b1)   (N x 512)
  dim3 gF1((N + 15) / 16, 4);
  gemm_wmma<DIM, 1><<<gF1, 256, 0, stream>>>(hbuf, W1, b1, t1, N, 4 * DIM);
  // 8) t2 = t1@W2 + b2        (N x 128, K=512)
  gemm_wmma<4 * DIM, 0><<<gN, 256, 0, stream>>>(t1, W2, b2, t2, N, DIM);
  // 9) out = LN(h + t2)
  ln_kernel<<<(N + 7) / 8, 256, 0, stream>>>(hbuf, t2, nullptr, g2, be2, outp, N);
}